// LenaBiTrans_23484881174859
// MI455X (gfx1250) — compile-verified
//
#include <hip/hip_runtime.h>
#include <hip/hip_bf16.h>

// ============================================================================
// LenaBiTrans forward, algebraically collapsed (exact):
//  * fe stage: LN over E=1 dim == emb_ln_b  ->  all tokens identical
//  * attention over identical tokens: ctx == V  (softmax rows sum to 1),
//    independent of scores / mask / rel_pe  -> both ALBERT layers reduce to
//    per-vector LN/FFN math; seq_out is one constant 128-vec z.
//  * only input-dependent work: head MLP on concat(z, station_e, day_e).
// One fused kernel: 8 blocks x 128 threads. Each block recomputes z (tiny),
// then runs a 16-row head tile with V_WMMA_F32_16X16X4_F32 (fp32 WMMA).
// ============================================================================

typedef __attribute__((ext_vector_type(2))) float v2f;
typedef __attribute__((ext_vector_type(8))) float v8f;

#define BATCH    128
#define TILE     16
#define HEAD_IN  160   // 128 (z) + 16 (station) + 16 (day)
#define TOPD     64

// ---- d_in index map -------------------------------------------------------
// setup_inputs() insertion order, params flattened as a JAX pytree
// (dict keys sorted alphabetically at each level, tuples in order):
//  0 tcf(unused) 1 product 2 day 3..10 cat_emb(unused) 11 day_emb
//  enc (sorted): 12 b1 13 b2 14 bk 15 bo 16 bq 17 bv 18 dist_emb 19 emb_ln_b
//   20 emb_ln_w 21 ln1_b 22 ln1_w 23 ln2_b 24 ln2_w 25 map_b 26 map_w
//   27 tok_emb 28 w1 29 w2 30 wk 31 wo 32 wq 33 wv
//  fe (sorted): 34 b1 35 b2 36 bk 37 bo 38 bq 39 bv 40 emb_ln_b 41 emb_ln_w
//   42 ln1_b 43 ln1_w 44 ln2_b 45 ln2_w 46 map_b 47 map_w 48 pos_emb
//   49 tok_emb 50 w1 51 w2 52 wk 53 wo 54 wq 55 wv
//  head (sorted): 56 b1 57 b2 58 b3 59 ln1_b 60 ln1_w 61 ln2_b 62 ln2_w
//   63 w1 64 w2 65 w3
//  66 station_emb
// ---------------------------------------------------------------------------

struct NetParams {
  const int *product, *day;
  const float *day_emb, *station_emb;
  // enc (64 -> 128)
  const float *e_b1,*e_b2,*e_bo,*e_bv,*e_embln_b,*e_embln_w;
  const float *e_ln1b,*e_ln1w,*e_ln2b,*e_ln2w,*e_mapb,*e_mapw,*e_tok;
  const float *e_w1,*e_w2,*e_wo,*e_wv;
  // fe (1 -> 64)
  const float *f_b1,*f_b2,*f_bo,*f_bv,*f_embln_b;
  const float *f_ln1b,*f_ln1w,*f_ln2b,*f_ln2w,*f_mapb,*f_mapw;
  const float *f_w1,*f_w2,*f_wo,*f_wv;
  // head
  const float *h_b1,*h_b2,*h_b3,*h_ln1b,*h_ln1w,*h_ln2b,*h_ln2w;
  const float *h_w1,*h_w2,*h_w3;
  float* out;
};

__device__ __forceinline__ float gelu_tanh(float x){
  float x3 = x*x*x;
  return 0.5f*x*(1.0f + tanhf(0.7978845608028654f*(x + 0.044715f*x3)));
}
__device__ __forceinline__ float gelu_erf(float x){
  return 0.5f*x*(1.0f + erff(x*0.70710678118654752f));
}

// LayerNorm (eps=1e-12) of shared vector v[0..n) in place; stats by thread 0.
__device__ void shared_ln(float* v, int n, const float* w, const float* b,
                          float* red, int tid){
  __syncthreads();
  if (tid == 0){
    float m = 0.f;
    for (int i = 0; i < n; ++i) m += v[i];
    m /= (float)n;
    float s = 0.f;
    for (int i = 0; i < n; ++i){ float d = v[i]-m; s += d*d; }
    red[0] = m;
    red[1] = 1.0f / sqrtf(s/(float)n + 1e-12f);
  }
  __syncthreads();
  if (tid < n) v[tid] = (v[tid]-red[0])*red[1]*w[tid] + b[tid];
  __syncthreads();
}

__global__ __launch_bounds__(128)
void lena_fused_kernel(NetParams p){
  __shared__ float s_c[64], s_v[64], s_t[64], s_h[64], s_feat[64];
  __shared__ float s_m[128], s_v0[128], s_x[128], s_h2[128], s_z[128];
  __shared__ float s_red[2];
  __shared__ float s_H1[TILE*HEAD_IN];   // 16 x 160 head input tile
  __shared__ float s_Y[TILE*TOPD];       // 16 x 64 pre-activation
  __shared__ float s_X[TILE*TOPD];       // 16 x 64 activation

  const int tid = threadIdx.x;

  // ===== Phase A: params-only trunk -> z[128] (recomputed per block) =====
  // fe token row c = emb_ln_b * map_w + map_b   (E=1 LN collapse)
  if (tid < 64)
    s_c[tid] = p.f_embln_b[0] * p.f_mapw[tid] + p.f_mapb[tid];
  __syncthreads();
  // ctx == vc = c@wv + bv (attention over identical tokens)
  if (tid < 64){
    float a = p.f_bv[tid];
    for (int k = 0; k < 64; ++k) a += s_c[k]*p.f_wv[k*64+tid];
    s_v[tid] = a;
  }
  __syncthreads();
  if (tid < 64){
    float a = s_c[tid] + p.f_bo[tid];
    for (int k = 0; k < 64; ++k) a += s_v[k]*p.f_wo[k*64+tid];
    s_t[tid] = a;
  }
  shared_ln(s_t, 64, p.f_ln1w, p.f_ln1b, s_red, tid);
  if (tid < 64){
    float a = p.f_b1[tid];
    for (int k = 0; k < 64; ++k) a += s_t[k]*p.f_w1[k*64+tid];
    s_h[tid] = gelu_tanh(a);
  }
  __syncthreads();
  if (tid < 64){
    float a = s_t[tid] + p.f_b2[tid];
    for (int k = 0; k < 64; ++k) a += s_h[k]*p.f_w2[k*64+tid];
    s_feat[tid] = a;
  }
  shared_ln(s_feat, 64, p.f_ln2w, p.f_ln2b, s_red, tid);

  // enc embedding path
  if (tid < 64) s_feat[tid] += p.e_tok[tid];
  shared_ln(s_feat, 64, p.e_embln_w, p.e_embln_b, s_red, tid);
  {
    float a = p.e_mapb[tid];
    for (int k = 0; k < 64; ++k) a += s_feat[k]*p.e_mapw[k*128+tid];
    s_m[tid] = a;
  }
  __syncthreads();
  {
    float a = p.e_bv[tid];
    for (int k = 0; k < 128; ++k) a += s_m[k]*p.e_wv[k*128+tid];
    s_v0[tid] = a;   // ctx == v0 again (identical tokens; mask/rel_pe vanish)
  }
  __syncthreads();
  {
    float a = s_m[tid] + p.e_bo[tid];
    for (int k = 0; k < 128; ++k) a += s_v0[k]*p.e_wo[k*128+tid];
    s_x[tid] = a;
  }
  shared_ln(s_x, 128, p.e_ln1w, p.e_ln1b, s_red, tid);
  {
    float a = p.e_b1[tid];
    for (int k = 0; k < 128; ++k) a += s_x[k]*p.e_w1[k*128+tid];
    s_h2[tid] = gelu_tanh(a);
  }
  __syncthreads();
  {
    float a = s_x[tid] + p.e_b2[tid];
    for (int k = 0; k < 128; ++k) a += s_h2[k]*p.e_w2[k*128+tid];
    s_z[tid] = a;
  }
  shared_ln(s_z, 128, p.e_ln2w, p.e_ln2b, s_red, tid);

  // ===== Phase B: build 16x160 head input tile [z | station_e | day_e] =====
  const int m0 = blockIdx.x * TILE;
  for (int idx = tid; idx < TILE*HEAD_IN; idx += 128){
    int r = idx / HEAD_IN;
    int c = idx - r*HEAD_IN;
    int row = m0 + r;
    float v;
    if      (c < 128) v = s_z[c];
    else if (c < 144) v = p.station_emb[p.product[row]*16 + (c-128)];
    else              v = p.day_emb[p.day[row]*16 + (c-144)];
    s_H1[idx] = v;
  }
  __syncthreads();

  // ===== Phase C: layer1  (16x160)@(160x64) via V_WMMA_F32_16X16X4_F32 =====
  const int lane = tid & 31;
  const int wid  = tid >> 5;     // wave -> 16-col N tile
  const int lh   = lane >> 4;    // lane half (K split)
  const int l15  = lane & 15;
  {
    const int n0 = wid * 16;
    v8f acc = {};
    for (int k0 = 0; k0 < HEAD_IN; k0 += 4){
      v2f a, b;
      const float* ar = &s_H1[l15*HEAD_IN + k0 + 2*lh];   // A: M=l15, K=k0+2*lh+{0,1}
      a.x = ar[0]; a.y = ar[1];
      const float* bc = &p.h_w1[(k0 + 2*lh)*TOPD + n0 + l15]; // B: K rows, N=l15
      b.x = bc[0]; b.y = bc[TOPD];
      acc = __builtin_amdgcn_wmma_f32_16x16x4_f32(
          false, a, false, b, (short)0, acc, false, false);
    }
    float bias = p.h_b1[n0 + l15];
#pragma unroll
    for (int r = 0; r < 8; ++r)
      s_Y[(r + 8*lh)*TOPD + n0 + l15] = acc[r] + bias;   // D: row = r + 8*half
  }
  __syncthreads();

  // ===== Phase D: per-row LN + exact GELU =====
  if (tid < TILE){
    const float* y = &s_Y[tid*TOPD];
    float m = 0.f;
    for (int i = 0; i < TOPD; ++i) m += y[i];
    m /= (float)TOPD;
    float s = 0.f;
    for (int i = 0; i < TOPD; ++i){ float d = y[i]-m; s += d*d; }
    float inv = 1.0f / sqrtf(s/(float)TOPD + 1e-12f);
    for (int i = 0; i < TOPD; ++i)
      s_X[tid*TOPD+i] = gelu_erf((y[i]-m)*inv*p.h_ln1w[i] + p.h_ln1b[i]);
  }
  __syncthreads();

  // ===== Phase E: layer2  (16x64)@(64x64) via WMMA =====
  {
    const int n0 = wid * 16;
    v8f acc = {};
    for (int k0 = 0; k0 < TOPD; k0 += 4){
      v2f a, b;
      const float* ar = &s_X[l15*TOPD + k0 + 2*lh];
      a.x = ar[0]; a.y = ar[1];
      const float* bc = &p.h_w2[(k0 + 2*lh)*TOPD + n0 + l15];
      b.x = bc[0]; b.y = bc[TOPD];
      acc = __builtin_amdgcn_wmma_f32_16x16x4_f32(
          false, a, false, b, (short)0, acc, false, false);
    }
    float bias = p.h_b2[n0 + l15];
#pragma unroll
    for (int r = 0; r < 8; ++r)
      s_Y[(r + 8*lh)*TOPD + n0 + l15] = acc[r] + bias;
  }
  __syncthreads();

  if (tid < TILE){
    const float* y = &s_Y[tid*TOPD];
    float m = 0.f;
    for (int i = 0; i < TOPD; ++i) m += y[i];
    m /= (float)TOPD;
    float s = 0.f;
    for (int i = 0; i < TOPD; ++i){ float d = y[i]-m; s += d*d; }
    float inv = 1.0f / sqrtf(s/(float)TOPD + 1e-12f);
    for (int i = 0; i < TOPD; ++i)
      s_X[tid*TOPD+i] = gelu_erf((y[i]-m)*inv*p.h_ln2w[i] + p.h_ln2b[i]);
  }
  __syncthreads();

  // ===== Phase F: final projection (64 -> 1) =====
  if (tid < TILE){
    float a = p.h_b3[0];
    const float* x = &s_X[tid*TOPD];
    for (int i = 0; i < TOPD; ++i) a += x[i]*p.h_w3[i];
    p.out[m0 + tid] = a;
  }
}

extern "C" void kernel_launch(void* const* d_in, const int* in_sizes, int n_in,
                              void* d_out, int out_size, void* d_ws, size_t ws_size,
                              hipStream_t stream){
  (void)in_sizes; (void)n_in; (void)d_ws; (void)ws_size; (void)out_size;
  NetParams p;
  p.product     = (const int*)d_in[1];
  p.day         = (const int*)d_in[2];
  p.day_emb     = (const float*)d_in[11];
  // enc
  p.e_b1   = (const float*)d_in[12];
  p.e_b2   = (const float*)d_in[13];
  p.e_bo   = (const float*)d_in[15];
  p.e_bv   = (const float*)d_in[17];
  p.e_embln_b = (const float*)d_in[19];
  p.e_embln_w = (const float*)d_in[20];
  p.e_ln1b = (const float*)d_in[21];
  p.e_ln1w = (const float*)d_in[22];
  p.e_ln2b = (const float*)d_in[23];
  p.e_ln2w = (const float*)d_in[24];
  p.e_mapb = (const float*)d_in[25];
  p.e_mapw = (const float*)d_in[26];
  p.e_tok  = (const float*)d_in[27];
  p.e_w1   = (const float*)d_in[28];
  p.e_w2   = (const float*)d_in[29];
  p.e_wo   = (const float*)d_in[31];
  p.e_wv   = (const float*)d_in[33];
  // fe
  p.f_b1   = (const float*)d_in[34];
  p.f_b2   = (const float*)d_in[35];
  p.f_bo   = (const float*)d_in[37];
  p.f_bv   = (const float*)d_in[39];
  p.f_embln_b = (const float*)d_in[40];
  p.f_ln1b = (const float*)d_in[42];
  p.f_ln1w = (const float*)d_in[43];
  p.f_ln2b = (const float*)d_in[44];
  p.f_ln2w = (const float*)d_in[45];
  p.f_mapb = (const float*)d_in[46];
  p.f_mapw = (const float*)d_in[47];
  p.f_w1   = (const float*)d_in[50];
  p.f_w2   = (const float*)d_in[51];
  p.f_wo   = (const float*)d_in[53];
  p.f_wv   = (const float*)d_in[55];
  // head
  p.h_b1   = (const float*)d_in[56];
  p.h_b2   = (const float*)d_in[57];
  p.h_b3   = (const float*)d_in[58];
  p.h_ln1b = (const float*)d_in[59];
  p.h_ln1w = (const float*)d_in[60];
  p.h_ln2b = (const float*)d_in[61];
  p.h_ln2w = (const float*)d_in[62];
  p.h_w1   = (const float*)d_in[63];
  p.h_w2   = (const float*)d_in[64];
  p.h_w3   = (const float*)d_in[65];
  p.station_emb = (const float*)d_in[66];
  p.out = (float*)d_out;

  lena_fused_kernel<<<dim3(BATCH/TILE), dim3(128), 0, stream>>>(p);
}